// TFN_38903813767517
// MI455X (gfx1250) — compile-verified
//
#include <hip/hip_runtime.h>
#include <math.h>

// ---------------- problem constants ----------------
#define N_NODES   100000
#define N_EDGES   1600000
#define HIDDEN    8
#define N_GRAPHS  64
#define N_TILES   (N_NODES / 16)   // 6250, exact

typedef float v2f __attribute__((ext_vector_type(2)));
typedef float v8f __attribute__((ext_vector_type(8)));

// softplus(n) - ln2 for n >= 0 (stable form)
__device__ __forceinline__ float ssp_act(float n) {
  return n + __logf(1.0f + __expf(-n)) - 0.6931471805599453f;
}

__device__ __forceinline__ float cutoff_w(float len) {
  float u = len * (1.0f / 1.5f);
  float u2 = u * u;
  float u6 = u2 * u2 * u2;
  float cut = 1.0f - 28.0f * u6 + 48.0f * u6 * u - 21.0f * u6 * u2;
  return (u < 1.0f) ? cut : 0.0f;
}

// ---------------- zero fill ----------------
__global__ void k_zero(float* __restrict__ p, int n) {
  int i = blockIdx.x * blockDim.x + threadIdx.x;
  if (i < n) p[i] = 0.0f;
}

// ============================================================
// Block 1 edge kernel: c_in = 1, paths (0,0,0),(0,1,1), out (N,4,8)
// one thread = (edge, out-channel d)
// ============================================================
__global__ __launch_bounds__(256) void k_edge1(
    const int* __restrict__ src, const int* __restrict__ dst,
    const float* __restrict__ xin, const float* __restrict__ sh,
    const float* __restrict__ elen,
    const float* __restrict__ wpw, const float* __restrict__ wpb, // (10,16),(16)
    float* __restrict__ agg) {
  __shared__ float sw[10 * 16 + 16];
  for (int i = threadIdx.x; i < 176; i += 256)
    sw[i] = (i < 160) ? wpw[i] : wpb[i - 160];
  __syncthreads();

  int idx = blockIdx.x * 256 + threadIdx.x;
  if (idx >= N_EDGES * 8) return;
  int e = idx >> 3, c = idx & 7;
  int s = src[e], d = dst[e];
  float len = elen[e];

  float lw0 = sw[160 + c], lw1 = sw[160 + 8 + c];
#pragma unroll
  for (int k = 0; k < 10; ++k) {
    float dk = len - (0.7f + (float)k * (1.0f / 9.0f));
    float ek = __expf(-50.0f * dk * dk);          // exp(-d^2 / (2*0.1^2))
    lw0 += ek * sw[k * 16 + c];
    lw1 += ek * sw[k * 16 + 8 + c];
  }
  float ew = cutoff_w(len);
  const float* she = sh + e * 9;
  float f = ew * xin[s];
  float* a = agg + d * 32 + c;
  atomicAdd(a + 0,  lw0 * f * she[0]);
  atomicAdd(a + 8,  lw1 * f * she[1]);
  atomicAdd(a + 16, lw1 * f * she[2]);
  atomicAdd(a + 24, lw1 * f * she[3]);
}

// ============================================================
// Block 2 edge kernel: channel-wise, 6 paths, (N,4,8) -> (N,4,8)
// ============================================================
__global__ __launch_bounds__(256) void k_edge2(
    const int* __restrict__ src, const int* __restrict__ dst,
    const float* __restrict__ hin, const float* __restrict__ sh,
    const float* __restrict__ elen,
    const float* __restrict__ wpw, const float* __restrict__ wpb, // (10,48),(48)
    float* __restrict__ agg) {
  __shared__ float sw[10 * 48 + 48];
  for (int i = threadIdx.x; i < 528; i += 256)
    sw[i] = (i < 480) ? wpw[i] : wpb[i - 480];
  __syncthreads();

  int idx = blockIdx.x * 256 + threadIdx.x;
  if (idx >= N_EDGES * 8) return;
  int e = idx >> 3, c = idx & 7;
  int s = src[e], d = dst[e];
  float len = elen[e];

  float lw[6];
#pragma unroll
  for (int p = 0; p < 6; ++p) lw[p] = sw[480 + p * 8 + c];
#pragma unroll
  for (int k = 0; k < 10; ++k) {
    float dk = len - (0.7f + (float)k * (1.0f / 9.0f));
    float ek = __expf(-50.0f * dk * dk);
#pragma unroll
    for (int p = 0; p < 6; ++p) lw[p] += ek * sw[k * 48 + p * 8 + c];
  }
  float ew = cutoff_w(len);

  const float* she = sh + e * 9;
  float s0 = she[0], s1 = she[1], s2 = she[2], s3 = she[3];
  float q0 = she[4], q1 = she[5], q2 = she[6], q3 = she[7], q4 = she[8];
  const float* hs = hin + s * 32 + c;
  float x0 = hs[0], v0 = hs[8], v1 = hs[16], v2 = hs[24];

  // real Clebsch-Gordan constants (derived analytically for this U convention)
  const float C110  = -0.57735026919f;  // -1/sqrt(3)   (1x1->0: -delta/sqrt3)
  const float C111  = -0.70710678119f;  // -1/sqrt(2)   (1x1->1: -eps/sqrt2)
  const float K121  =  0.54772255751f;  // sqrt(3/10)
  const float K121a =  0.31622776601f;  // sqrt(1/10)
  const float K121b =  0.63245553203f;  // 2*sqrt(1/10)

  float out0 = lw[0] * (x0 * s0) + lw[3] * C110 * (v0 * s1 + v1 * s2 + v2 * s3);
  // cross product (x-vec cross sh-vec) for path (1,1,1)
  float cr0 = v1 * s3 - v2 * s2;
  float cr1 = v2 * s1 - v0 * s3;
  float cr2 = v0 * s2 - v1 * s1;
  // path (1,2,1) contraction with cg121 = -sqrt(3/10)*S
  float m0 = -K121 * (v2 * q0 + v1 * q1) + K121a * (v0 * q2) + K121 * (v0 * q4);
  float m1 = -K121 * (v0 * q1) - K121b * (v1 * q2) - K121 * (v2 * q3);
  float m2 = -K121 * (v0 * q0) + K121a * (v2 * q2) - K121 * (v1 * q3) - K121 * (v2 * q4);

  float out1 = lw[1] * (x0 * s1) + lw[2] * (v0 * s0) + lw[4] * C111 * cr0 + lw[5] * m0;
  float out2 = lw[1] * (x0 * s2) + lw[2] * (v1 * s0) + lw[4] * C111 * cr1 + lw[5] * m1;
  float out3 = lw[1] * (x0 * s3) + lw[2] * (v2 * s0) + lw[4] * C111 * cr2 + lw[5] * m2;

  float* a = agg + d * 32 + c;
  atomicAdd(a + 0,  ew * out0);
  atomicAdd(a + 8,  ew * out1);
  atomicAdd(a + 16, ew * out2);
  atomicAdd(a + 24, ew * out3);
}

// ============================================================
// Block 3 edge kernel: channel-wise, paths (0,0,0),(1,1,0), out (N,1,8)
// ============================================================
__global__ __launch_bounds__(256) void k_edge3(
    const int* __restrict__ src, const int* __restrict__ dst,
    const float* __restrict__ hin, const float* __restrict__ sh,
    const float* __restrict__ elen,
    const float* __restrict__ wpw, const float* __restrict__ wpb, // (10,16),(16)
    float* __restrict__ agg) {
  __shared__ float sw[10 * 16 + 16];
  for (int i = threadIdx.x; i < 176; i += 256)
    sw[i] = (i < 160) ? wpw[i] : wpb[i - 160];
  __syncthreads();

  int idx = blockIdx.x * 256 + threadIdx.x;
  if (idx >= N_EDGES * 8) return;
  int e = idx >> 3, c = idx & 7;
  int s = src[e], d = dst[e];
  float len = elen[e];

  float lw0 = sw[160 + c], lw1 = sw[160 + 8 + c];
#pragma unroll
  for (int k = 0; k < 10; ++k) {
    float dk = len - (0.7f + (float)k * (1.0f / 9.0f));
    float ek = __expf(-50.0f * dk * dk);
    lw0 += ek * sw[k * 16 + c];
    lw1 += ek * sw[k * 16 + 8 + c];
  }
  float ew = cutoff_w(len);

  const float* she = sh + e * 9;
  const float* hs = hin + s * 32 + c;
  float x0 = hs[0], v0 = hs[8], v1 = hs[16], v2 = hs[24];
  const float C110 = -0.57735026919f;
  float out0 = lw0 * (x0 * she[0]) +
               lw1 * C110 * (v0 * she[1] + v1 * she[2] + v2 * she[3]);
  atomicAdd(agg + d * 8 + c, ew * out0);
}

// ============================================================
// Node kernel blocks 1/2: self-interaction via V_WMMA_F32_16X16X4_F32
// (16-node tile) x (8x8 weight, col-padded to 16), K=8 as two chained
// K=4 WMMAs.  Fused norm-gated activation in registers, no shuffles.
// si layout: (2,8,8); order 0 -> si[0], orders 1..3 -> si[1].
// ============================================================
__global__ __launch_bounds__(256) void k_node12(
    const float* __restrict__ agg, const float* __restrict__ si,
    float* __restrict__ hout) {
  int wave = threadIdx.x >> 5;
  int tile = blockIdx.x * 8 + wave;
  if (tile >= N_TILES) return;               // wave-uniform: EXEC stays full
  int lane = threadIdx.x & 31;
  int half = lane >> 4;                      // K-group select
  int col  = lane & 15;                      // A row (M) / B,D column (N)
  int k0   = half * 2;
  int nbase = tile * 16;

  // B operands (8x8 weight, columns 8..15 zero-padded)
  float b00 = 0, b01 = 0, b02 = 0, b03 = 0, b10 = 0, b11 = 0, b12 = 0, b13 = 0;
  if (col < 8) {
    const float* W0 = si;        // (c,d) row-major
    const float* W1 = si + 64;
    b00 = W0[(k0 + 0) * 8 + col]; b01 = W0[(k0 + 1) * 8 + col];
    b02 = W0[(k0 + 4) * 8 + col]; b03 = W0[(k0 + 5) * 8 + col];
    b10 = W1[(k0 + 0) * 8 + col]; b11 = W1[(k0 + 1) * 8 + col];
    b12 = W1[(k0 + 4) * 8 + col]; b13 = W1[(k0 + 5) * 8 + col];
  }
  v2f bLo0 = {b00, b01}, bHi0 = {b02, b03};
  v2f bLo1 = {b10, b11}, bHi1 = {b12, b13};

  v8f dacc[4];
  const float* arow = agg + (nbase + col) * 32;   // A row M = lane%16
#pragma unroll
  for (int m = 0; m < 4; ++m) {
    const float* ar = arow + m * 8;
    v2f aLo = {ar[k0 + 0], ar[k0 + 1]};
    v2f aHi = {ar[k0 + 4], ar[k0 + 5]};
    v2f bl = (m == 0) ? bLo0 : bLo1;
    v2f bh = (m == 0) ? bHi0 : bHi1;
    v8f cz = {0.f, 0.f, 0.f, 0.f, 0.f, 0.f, 0.f, 0.f};
    v8f t = __builtin_amdgcn_wmma_f32_16x16x4_f32(false, aLo, false, bl,
                                                  (short)0, cz, false, false);
    dacc[m] = __builtin_amdgcn_wmma_f32_16x16x4_f32(false, aHi, false, bh,
                                                    (short)0, t, false, false);
  }

  // fused norm_act: for fixed (node, channel) all 4 order values live in
  // the same lane, same vector index v across dacc[0..3].
  if (col < 8) {
#pragma unroll
    for (int v = 0; v < 8; ++v) {
      int node = nbase + v + 8 * half;           // D row M = v + 8*half
      float h0 = dacc[0][v], h1 = dacc[1][v], h2 = dacc[2][v], h3 = dacc[3][v];
      float n0 = sqrtf(h0 * h0 + 1e-12f);
      float g0 = ssp_act(n0) / n0;
      float n1 = sqrtf(h1 * h1 + h2 * h2 + h3 * h3 + 1e-12f);
      float g1 = ssp_act(n1) / n1;
      float* o = hout + node * 32 + col;
      o[0]  = h0 * g0;
      o[8]  = h1 * g1;
      o[16] = h2 * g1;
      o[24] = h3 * g1;
    }
  }
}

// ============================================================
// Node kernel block 3: WMMA self-interaction + SiLU + per-graph pooling.
// batch is sorted -> stage per-graph sums in LDS, flush nonzeros.
// ============================================================
__global__ __launch_bounds__(256) void k_node3(
    const float* __restrict__ agg,   // (N,8)
    const float* __restrict__ si,    // (8,8)
    const int* __restrict__ batch,
    float* __restrict__ g) {         // (64,8)
  __shared__ float gacc[N_GRAPHS * 8];
  for (int i = threadIdx.x; i < N_GRAPHS * 8; i += 256) gacc[i] = 0.0f;
  __syncthreads();

  int wave = threadIdx.x >> 5;
  int tile = blockIdx.x * 8 + wave;
  int lane = threadIdx.x & 31;
  int half = lane >> 4, col = lane & 15, k0 = half * 2;

  if (tile < N_TILES) {                        // wave-uniform
    int nbase = tile * 16;
    float b0 = 0, b1 = 0, b2 = 0, b3 = 0;
    if (col < 8) {
      b0 = si[(k0 + 0) * 8 + col]; b1 = si[(k0 + 1) * 8 + col];
      b2 = si[(k0 + 4) * 8 + col]; b3 = si[(k0 + 5) * 8 + col];
    }
    v2f bLo = {b0, b1}, bHi = {b2, b3};
    const float* ar = agg + (nbase + col) * 8;
    v2f aLo = {ar[k0 + 0], ar[k0 + 1]};
    v2f aHi = {ar[k0 + 4], ar[k0 + 5]};
    v8f cz = {0.f, 0.f, 0.f, 0.f, 0.f, 0.f, 0.f, 0.f};
    v8f t  = __builtin_amdgcn_wmma_f32_16x16x4_f32(false, aLo, false, bLo,
                                                   (short)0, cz, false, false);
    v8f dd = __builtin_amdgcn_wmma_f32_16x16x4_f32(false, aHi, false, bHi,
                                                   (short)0, t, false, false);
    if (col < 8) {
#pragma unroll
      for (int v = 0; v < 8; ++v) {
        int node = nbase + v + 8 * half;
        float val = dd[v];
        float sl = val / (1.0f + __expf(-val));  // silu
        int b = batch[node];
        atomicAdd(&gacc[b * 8 + col], sl);
      }
    }
  }
  __syncthreads();
  for (int i = threadIdx.x; i < N_GRAPHS * 8; i += 256) {
    float val = gacc[i];
    if (val != 0.0f) atomicAdd(&g[i], val);
  }
}

// ============================================================
// Head: logits = g @ out_w + out_b ; softmax over 8
// ============================================================
__global__ void k_head(const float* __restrict__ g, const float* __restrict__ ow,
                       const float* __restrict__ ob, float* __restrict__ out) {
  int gi = blockIdx.x * blockDim.x + threadIdx.x;
  if (gi >= N_GRAPHS) return;
  float gv[8], lg[8];
#pragma unroll
  for (int c = 0; c < 8; ++c) gv[c] = g[gi * 8 + c];
  float mx = -1e30f;
#pragma unroll
  for (int d = 0; d < 8; ++d) {
    float acc = ob[d];
#pragma unroll
    for (int c = 0; c < 8; ++c) acc += gv[c] * ow[c * 8 + d];
    lg[d] = acc;
    mx = fmaxf(mx, acc);
  }
  float sum = 0.0f;
#pragma unroll
  for (int d = 0; d < 8; ++d) { lg[d] = __expf(lg[d] - mx); sum += lg[d]; }
  float inv = 1.0f / sum;
#pragma unroll
  for (int d = 0; d < 8; ++d) out[gi * 8 + d] = lg[d] * inv;
}

// ============================================================
// launch
// ============================================================
extern "C" void kernel_launch(void* const* d_in, const int* in_sizes, int n_in,
                              void* d_out, int out_size, void* d_ws, size_t ws_size,
                              hipStream_t stream) {
  const float* x     = (const float*)d_in[0];
  const int*   eidx  = (const int*)d_in[1];
  const float* esh   = (const float*)d_in[2];
  const float* elen  = (const float*)d_in[3];
  const int*   batch = (const int*)d_in[4];
  const float* wp1w  = (const float*)d_in[5];
  const float* wp1b  = (const float*)d_in[6];
  const float* wp2w  = (const float*)d_in[7];
  const float* wp2b  = (const float*)d_in[8];
  const float* wp3w  = (const float*)d_in[9];
  const float* wp3b  = (const float*)d_in[10];
  const float* si1   = (const float*)d_in[11];
  const float* si2   = (const float*)d_in[12];
  const float* si3   = (const float*)d_in[13];
  const float* outw  = (const float*)d_in[14];
  const float* outb  = (const float*)d_in[15];
  float* out = (float*)d_out;

  const int* src = eidx;
  const int* dst = eidx + N_EDGES;

  // workspace layout (needs ~25.6 MB)
  float* aggbuf = (float*)d_ws;                       // N*32 floats
  float* hbuf   = aggbuf + (size_t)N_NODES * 32;      // N*32 floats
  float* gbuf   = hbuf + (size_t)N_NODES * 32;        // 64*8 floats

  const int NF    = N_NODES * 32;
  const int zgrid = (NF + 255) / 256;
  const int egrid = (N_EDGES * 8 + 255) / 256;        // 50000
  const int ngrid = (N_TILES + 7) / 8;                // 782

  // ---- block 1 ----
  k_zero<<<zgrid, 256, 0, stream>>>(aggbuf, NF);
  k_edge1<<<egrid, 256, 0, stream>>>(src, dst, x, esh, elen, wp1w, wp1b, aggbuf);
  k_node12<<<ngrid, 256, 0, stream>>>(aggbuf, si1, hbuf);

  // ---- block 2 ----
  k_zero<<<zgrid, 256, 0, stream>>>(aggbuf, NF);
  k_edge2<<<egrid, 256, 0, stream>>>(src, dst, hbuf, esh, elen, wp2w, wp2b, aggbuf);
  k_node12<<<ngrid, 256, 0, stream>>>(aggbuf, si2, hbuf);

  // ---- block 3 ----
  k_zero<<<(N_NODES * 8 + 255) / 256, 256, 0, stream>>>(aggbuf, N_NODES * 8);
  k_zero<<<(N_GRAPHS * 8 + 255) / 256, 256, 0, stream>>>(gbuf, N_GRAPHS * 8);
  k_edge3<<<egrid, 256, 0, stream>>>(src, dst, hbuf, esh, elen, wp3w, wp3b, aggbuf);
  k_node3<<<ngrid, 256, 0, stream>>>(aggbuf, si3, batch, gbuf);

  // ---- head ----
  k_head<<<1, 64, 0, stream>>>(gbuf, outw, outb, out);
}